// CustomProposalLayer_32091995636088
// MI455X (gfx1250) — compile-verified
//
#include <hip/hip_runtime.h>
#include <math.h>

// CDNA5 / gfx1250 proposal layer: selection-first pipeline, WMMA f32 Gram-diagonal
// for embedding L2 norms.

typedef float v2f __attribute__((ext_vector_type(2)));
typedef float v8f __attribute__((ext_vector_type(8)));

#define BATCH  16
#define NANCH  4
#define KLVL   300
#define NCAND  900     // 3 levels * 300
#define MAXP   500
#define EMB    208
#define DDIM   213     // 4 box + 1 conf + 208 emb
#define IOU_T  0.45f
#define TOPK_T 1024

__constant__ float ANCH[3][4][2] = {
  {{8.f,24.f},{11.f,34.f},{16.f,48.f},{23.f,68.f}},
  {{32.f,96.f},{45.f,135.f},{64.f,192.f},{90.f,271.f}},
  {{128.f,384.f},{180.f,540.f},{256.f,640.f},{512.f,640.f}}};
__constant__ int   HH[3]  = {76,38,19};
__constant__ float STR[3] = {8.f,16.f,32.f};

// ---------------------------------------------------------------- zero output
__global__ void zero_kernel(float* __restrict__ out, int n) {
  for (int i = blockIdx.x*blockDim.x + threadIdx.x; i < n; i += gridDim.x*blockDim.x)
    out[i] = 0.0f;
}

// ------------------------------------------------- per-level masked confidence
// conf = sigmoid(p5 - p4); masked to 0 when <= 0.5 (used for ranking; selected
// entries always have s > 0.5 so masked == unmasked for kept proposals).
__global__ void score_kernel(const float* __restrict__ pred,
                             float* __restrict__ conf, int perImg) {
  int i = blockIdx.x*blockDim.x + threadIdx.x;
  if (i >= BATCH*perImg) return;
  const float* p = pred + (size_t)i*6;
  float s = 1.0f / (1.0f + expf(p[4] - p[5]));
  conf[i] = (s > 0.5f) ? s : 0.0f;
}

// --------------------------------------- top-300 per (image, level), stable ties
// Iterated strictly-decreasing max; order key = (score desc, index asc), which
// matches jax.lax.top_k on the masked scores.
__global__ __launch_bounds__(TOPK_T)
void topk_kernel(const float* __restrict__ conf, int N, int lvl,
                 float* __restrict__ selS, int* __restrict__ selI) {
  int b = blockIdx.x;
  int t = threadIdx.x;
  const float* c = conf + (size_t)b*N;

  float v[23];                       // max N = 23104 -> ceil(23104/1024) = 23
  #pragma unroll
  for (int j = 0; j < 23; ++j) {
    int i = t + j*TOPK_T;
    v[j] = (i < N) ? c[i] : -1.0f;
  }

  __shared__ float sv[TOPK_T];
  __shared__ int   si[TOPK_T];
  __shared__ float bS;
  __shared__ int   bI;

  float pS = 3.0e38f; int pI = -1;
  float* outS = selS + (b*3 + lvl)*KLVL;
  int*   outI = selI + (b*3 + lvl)*KLVL;

  for (int k = 0; k < KLVL; ++k) {
    float ls = -2.0f; int li = 0x7fffffff;
    #pragma unroll
    for (int j = 0; j < 23; ++j) {
      int i = t + j*TOPK_T;
      float s = v[j];
      bool elig   = (s < pS) || (s == pS && i > pI);
      bool better = elig && ((s > ls) || (s == ls && i < li));
      if (better) { ls = s; li = i; }
    }
    sv[t] = ls; si[t] = li;
    __syncthreads();
    for (int off = TOPK_T/2; off > 0; off >>= 1) {
      if (t < off) {
        float s2 = sv[t+off]; int i2 = si[t+off];
        if (s2 > sv[t] || (s2 == sv[t] && i2 < si[t])) { sv[t] = s2; si[t] = i2; }
      }
      __syncthreads();
    }
    if (t == 0) {
      bS = sv[0]; bI = si[0];
      outS[k] = (sv[0] > 0.0f) ? sv[0] : 0.0f;
      outI[k] = (sv[0] > 0.0f) ? si[0] : -1;
    }
    __syncthreads();
    pS = bS; pI = bI;
    __syncthreads();
  }
}

// --------------------------------------------------- decode boxes for selected
__global__ void decode_kernel(const float* __restrict__ pred0,
                              const float* __restrict__ pred1,
                              const float* __restrict__ pred2,
                              const float* __restrict__ selS,
                              const int*   __restrict__ selI,
                              float* __restrict__ boxb) {
  int i = blockIdx.x*blockDim.x + threadIdx.x;
  if (i >= BATCH*NCAND) return;
  int b = i / NCAND, j = i % NCAND;
  int lvl = j / KLVL, k = j % KLVL;
  float s = selS[(b*3 + lvl)*KLVL + k];
  int   n = selI[(b*3 + lvl)*KLVL + k];
  float4 box = make_float4(0.f, 0.f, 0.f, 0.f);
  if (s > 0.0f && n >= 0) {
    int H = HH[lvl], HW = H*H;
    int a = n / HW, rem = n - a*HW, h = rem / H, w = rem - h*H;
    const float* pr = (lvl == 0) ? pred0 : ((lvl == 1) ? pred1 : pred2);
    const float* p = pr + ((size_t)((b*NANCH + a)*H + h)*H + w)*6;
    float st = STR[lvl];
    float aw = ANCH[lvl][a][0]/st, ah = ANCH[lvl][a][1]/st;
    float x  = p[0]*aw + (float)w;
    float y  = p[1]*ah + (float)h;
    float bw = expf(p[2])*aw;
    float bh = expf(p[3])*ah;
    float sc = st / 608.0f;
    box.x = fminf(fmaxf((x - 0.5f*bw)*sc, 0.f), 1.f);
    box.y = fminf(fmaxf((y - 0.5f*bh)*sc, 0.f), 1.f);
    box.z = fminf(fmaxf((x + 0.5f*bw)*sc, 0.f), 1.f);
    box.w = fminf(fmaxf((y + 0.5f*bh)*sc, 0.f), 1.f);
  }
  *(float4*)(boxb + (size_t)i*4) = box;
}

// -------------------------------------------------------- per-image greedy NMS
__global__ __launch_bounds__(TOPK_T)
void nms_kernel(const float* __restrict__ selS, const int* __restrict__ selI,
                const float* __restrict__ boxb, float* __restrict__ out,
                int* __restrict__ map) {
  int b = blockIdx.x, t = threadIdx.x;
  __shared__ float us[NCAND];
  __shared__ float bx[NCAND], by[NCAND], bX[NCAND], bY[NCAND];
  __shared__ float ss[NCAND], ar[NCAND];
  __shared__ int   src[NCAND], kp[NCAND], dst[NCAND];

  for (int s0 = t; s0 < MAXP; s0 += blockDim.x) map[b*MAXP + s0] = -1;

  for (int j = t; j < NCAND; j += blockDim.x) {
    int lvl = j / KLVL, k = j % KLVL;
    us[j] = selS[(b*3 + lvl)*KLVL + k];
  }
  __syncthreads();

  // stable descending rank (ties -> lower original index first), scatter sorted
  for (int j = t; j < NCAND; j += blockDim.x) {
    float s = us[j];
    int r = 0;
    for (int m = 0; m < NCAND; ++m) {
      float sm = us[m];
      r += (sm > s || (sm == s && m < j)) ? 1 : 0;
    }
    int lvl = j / KLVL, k = j % KLVL;
    int n   = selI[(b*3 + lvl)*KLVL + k];
    const float* bb = boxb + ((size_t)(b*NCAND + j))*4;
    float x1 = bb[0], y1 = bb[1], x2 = bb[2], y2 = bb[3];
    bx[r] = x1; by[r] = y1; bX[r] = x2; bY[r] = y2;
    ss[r] = s;
    ar[r] = (x2 - x1)*(y2 - y1);
    src[r] = ((unsigned)lvl << 28) | ((unsigned)n & 0x0FFFFFFF);
    kp[r] = (s > 0.0f) ? 1 : 0;
  }
  __syncthreads();

  // greedy suppression in sorted order
  for (int i = 0; i < NCAND; ++i) {
    __syncthreads();
    if (!kp[i]) continue;                 // uniform branch (shared value)
    float ix1 = bx[i], iy1 = by[i], ix2 = bX[i], iy2 = bY[i], ia = ar[i];
    for (int j = i + 1 + t; j < NCAND; j += blockDim.x) {
      float xx1 = fmaxf(ix1, bx[j]);
      float yy1 = fmaxf(iy1, by[j]);
      float xx2 = fminf(ix2, bX[j]);
      float yy2 = fminf(iy2, bY[j]);
      float inter = fmaxf(xx2 - xx1, 0.f) * fmaxf(yy2 - yy1, 0.f);
      float iou = inter / (ia + ar[j] - inter + 1e-9f);
      if (iou > IOU_T) kp[j] = 0;
    }
  }
  __syncthreads();

  if (t == 0) {
    int cnt = 0;
    for (int i = 0; i < NCAND; ++i) {
      if (kp[i] && ss[i] > 0.0f) { dst[i] = (cnt < MAXP) ? cnt : -1; ++cnt; }
      else dst[i] = -1;
    }
  }
  __syncthreads();

  for (int i = t; i < NCAND; i += blockDim.x) {
    int d = dst[i];
    if (d >= 0) {
      float* o = out + (size_t)(b*MAXP + d)*DDIM;
      o[0] = bx[i]; o[1] = by[i]; o[2] = bX[i]; o[3] = bY[i]; o[4] = ss[i];
      map[b*MAXP + d] = src[i];
    }
  }
}

// ---------------- embedding gather + L2 normalize via WMMA f32 Gram diagonal
// One wave32 per group of 16 output slots. Tile = 16 rows x 208 f32 in LDS.
// norms = diag(E * E^T) accumulated with 52x V_WMMA_F32_16X16X4_F32.
__global__ __launch_bounds__(32)
void emb_kernel(const float* __restrict__ emb0, const float* __restrict__ emb1,
                const float* __restrict__ emb2, const int* __restrict__ map,
                float* __restrict__ out) {
  int bid = blockIdx.x;
  int b = bid >> 5;          // 32 groups of 16 slots per image (covers 500)
  int g = bid & 31;
  int lane = threadIdx.x;
  int r    = lane & 15;      // row within 16-slot tile (also WMMA M index)
  int half = lane >> 4;

  __shared__ float tile[16][EMB];
  __shared__ float diag[8][32];
  __shared__ float nrm[16];

  int slot = g*16 + r;
  int srcv = (slot < MAXP) ? map[b*MAXP + slot] : -1;
  int colbase = half * (EMB/2);

  if (srcv >= 0) {
    int lvl = ((unsigned)srcv) >> 28;
    int n   = srcv & 0x0FFFFFFF;
    int H = HH[lvl], HW = H*H;
    int rem = n % HW, h = rem / H, w = rem - h*H;
    const float* e = ((lvl == 0) ? emb0 : ((lvl == 1) ? emb1 : emb2))
                     + ((size_t)((b*H + h)*H + w))*EMB + colbase;
    for (int c = 0; c < EMB/2; c += 4)
      *(float4*)&tile[r][colbase + c] = *(const float4*)(e + c);
  } else {
    for (int c = 0; c < EMB/2; c += 4)
      *(float4*)&tile[r][colbase + c] = make_float4(0.f, 0.f, 0.f, 0.f);
  }
  __syncthreads();

  // A 16x4 f32 layout: lanes 0-15 hold K0/K1, lanes 16-31 hold K2/K3 (M = lane%16).
  // B = A^T has the symmetric striping, so the same register pair feeds both.
  v8f acc = {};
  int kofs = half ? 2 : 0;
  for (int k4 = 0; k4 < EMB; k4 += 4) {
    v2f a;
    a.x = tile[r][k4 + kofs];
    a.y = tile[r][k4 + kofs + 1];
    acc = __builtin_amdgcn_wmma_f32_16x16x4_f32(false, a, false, a,
                                                (short)0, acc, false, false);
  }
  #pragma unroll
  for (int vv = 0; vv < 8; ++vv) diag[vv][lane] = acc[vv];
  __syncthreads();

  if (lane < 16) {
    float nv = (lane < 8) ? diag[lane][lane] : diag[lane - 8][lane + 16];
    nrm[lane] = rsqrtf(fmaxf(nv, 1e-12f));
  }
  __syncthreads();

  if (slot < MAXP) {
    float sc = nrm[r];
    float* o = out + (size_t)(b*MAXP + slot)*DDIM + 5 + colbase;
    for (int c = 0; c < EMB/2; ++c) o[c] = tile[r][colbase + c] * sc;
  }
}

// ----------------------------------------------------------------------------
extern "C" void kernel_launch(void* const* d_in, const int* in_sizes, int n_in,
                              void* d_out, int out_size, void* d_ws, size_t ws_size,
                              hipStream_t stream) {
  // setup_inputs() dict order: pred0, emb0, pred1, emb1, pred2, emb2
  const float* pred0 = (const float*)d_in[0];
  const float* emb0  = (const float*)d_in[1];
  const float* pred1 = (const float*)d_in[2];
  const float* emb1  = (const float*)d_in[3];
  const float* pred2 = (const float*)d_in[4];
  const float* emb2  = (const float*)d_in[5];
  float* out = (float*)d_out;

  const int AHW0 = NANCH*76*76;   // 23104
  const int AHW1 = NANCH*38*38;   // 5776
  const int AHW2 = NANCH*19*19;   // 1444

  float* ws    = (float*)d_ws;
  float* conf0 = ws;
  float* conf1 = conf0 + (size_t)BATCH*AHW0;
  float* conf2 = conf1 + (size_t)BATCH*AHW1;
  float* selS  = conf2 + (size_t)BATCH*AHW2;
  int*   selI  = (int*)(selS + (size_t)BATCH*NCAND);
  float* boxb  = (float*)(selI + (size_t)BATCH*NCAND);
  int*   map   = (int*)(boxb + (size_t)BATCH*NCAND*4);

  zero_kernel<<<512, 256, 0, stream>>>(out, out_size);

  score_kernel<<<(BATCH*AHW0 + 255)/256, 256, 0, stream>>>(pred0, conf0, AHW0);
  score_kernel<<<(BATCH*AHW1 + 255)/256, 256, 0, stream>>>(pred1, conf1, AHW1);
  score_kernel<<<(BATCH*AHW2 + 255)/256, 256, 0, stream>>>(pred2, conf2, AHW2);

  topk_kernel<<<BATCH, TOPK_T, 0, stream>>>(conf0, AHW0, 0, selS, selI);
  topk_kernel<<<BATCH, TOPK_T, 0, stream>>>(conf1, AHW1, 1, selS, selI);
  topk_kernel<<<BATCH, TOPK_T, 0, stream>>>(conf2, AHW2, 2, selS, selI);

  decode_kernel<<<(BATCH*NCAND + 255)/256, 256, 0, stream>>>(
      pred0, pred1, pred2, selS, selI, boxb);

  nms_kernel<<<BATCH, TOPK_T, 0, stream>>>(selS, selI, boxb, out, map);

  emb_kernel<<<BATCH*32, 32, 0, stream>>>(emb0, emb1, emb2, map, out);
}